// CovPool_85839216377861
// MI455X (gfx1250) — compile-verified
//
#include <hip/hip_runtime.h>

// CovPool on MI455X (gfx1250, wave32, WMMA):
//   cov_b = (X_b^T X_b - N mu mu^T)/(N-1) + lam*I, rows lex-sorted ascending.
// Stage 1: batched partial Gram via V_WMMA_F32_16X16X4_F32 with double-buffered
//          GLOBAL_LOAD_ASYNC_TO_LDS_B128 staging (ASYNCcnt pipelined).
// Stage 2: deterministic fixed-order reduction + mean correction + lex sort.

typedef __attribute__((ext_vector_type(2))) float v2f;
typedef __attribute__((ext_vector_type(8))) float v8f;
typedef __attribute__((ext_vector_type(4))) int   v4i;

#define BATCH     32
#define NPTS      8192
#define DDIM      128
#define LAM       0.01f
#define KSLICES   8                    // K-split per batch -> 256 workgroups
#define KSLICE    (NPTS / KSLICES)     // 1024
#define CHUNK     32                   // rows staged in LDS per pipeline stage
#define NCHUNK    (KSLICE / CHUNK)     // 32
#define LDS_STRIDE 136                 // 128 + 8 pad: (2*136)%64==16 -> half-wave
                                       // bank groups disjoint for frag reads
#define CP_PER_THREAD ((CHUNK * DDIM) / 4 / 256)   // 4 float4 copies per thread

#if defined(__has_builtin)
#if __has_builtin(__builtin_amdgcn_global_load_async_to_lds_b128)
#define HAVE_ASYNC_LDS 1
#endif
#endif

#ifdef HAVE_ASYNC_LDS
#pragma message("gfx1250 path: GLOBAL_LOAD_ASYNC_TO_LDS_B128 double-buffered staging")
// Builtin signature (per hipcc diagnostic): param 0 is AS(1) int4* ("__device__"),
// so: (global v4i*, lds v4i*, imm offset, imm cpol).
typedef __attribute__((address_space(1))) v4i g_v4i;
typedef __attribute__((address_space(3))) v4i l_v4i;

__device__ __forceinline__ void async_copy16(const void* g, void* l)
{
    // Generic LDS address truncates to the AS(3) offset; global generic == AS(1).
    __builtin_amdgcn_global_load_async_to_lds_b128(
        (g_v4i*)(unsigned long long)g,
        (l_v4i*)(unsigned int)(unsigned long long)l,
        /*offset=*/0, /*cpol=*/0);
}
#else
#pragma message("gfx1250 path: fallback register staging (async builtin unavailable)")
#endif

// ---------------------------------------------------------------------------
// Stage 1: partial Gram (128x128) + partial column sums for one (batch, slice)
// block = 256 threads = 8 waves; wave w owns output rows [16w, 16w+16).
// ---------------------------------------------------------------------------
__global__ __launch_bounds__(256) void gram_partial(
    const float* __restrict__ x,     // (B, N, D) fp32
    float* __restrict__ gram_ws,     // (B, KSLICES, 128, 128)
    float* __restrict__ sum_ws)      // (B, KSLICES, 128)
{
    const int bs = blockIdx.x;
    const int b  = bs / KSLICES;
    const int s  = bs % KSLICES;
    const float* __restrict__ xb =
        x + ((size_t)b * NPTS + (size_t)s * KSLICE) * DDIM;

    __shared__ float lds[2][CHUNK * LDS_STRIDE];

    const int tid  = threadIdx.x;
    const int wave = tid >> 5;
    const int lane = tid & 31;
    const int l15  = lane & 15;
    const int koff = (lane >> 4) << 1;   // 0 for lanes 0-15, 2 for lanes 16-31
    const int mrow = wave * 16;          // this wave's output row strip

    v8f acc[8] = {};                     // 8 tiles of 16x16 f32 accumulators

    // column-sum accumulation (for the mean correction)
    float csum = 0.f;
    const int sc = tid & (DDIM - 1);           // column owned by this thread
    const int sh = (tid >> 7) * (CHUNK / 2);   // row half of each chunk

#ifdef HAVE_ASYNC_LDS
    // ---- prologue: issue chunk 0 into buffer 0 ----
    {
        const float* __restrict__ src = xb;
        #pragma unroll
        for (int j = 0; j < CP_PER_THREAD; ++j) {
            const int idx = j * 256 + tid;            // float4 index in chunk
            const int r   = idx >> 5;
            const int c4  = idx & 31;
            async_copy16(src + (size_t)idx * 4, &lds[0][r * LDS_STRIDE + c4 * 4]);
        }
    }
#endif

    for (int ch = 0; ch < NCHUNK; ++ch) {
        const int cur = ch & 1;

#ifdef HAVE_ASYNC_LDS
        if (ch + 1 < NCHUNK) {
            // issue next chunk into the other buffer (safe: it was fully
            // consumed before the trailing barrier of iteration ch-1)
            const float* __restrict__ src = xb + (size_t)(ch + 1) * CHUNK * DDIM;
            #pragma unroll
            for (int j = 0; j < CP_PER_THREAD; ++j) {
                const int idx = j * 256 + tid;
                const int r   = idx >> 5;
                const int c4  = idx & 31;
                async_copy16(src + (size_t)idx * 4,
                             &lds[cur ^ 1][r * LDS_STRIDE + c4 * 4]);
            }
            // async loads retire in order: leaving the newest 4 outstanding
            // guarantees this wave's 4 current-chunk copies are in LDS.
            asm volatile("s_wait_asynccnt 0x4" ::: "memory");
        } else {
            asm volatile("s_wait_asynccnt 0x0" ::: "memory");
        }
        __syncthreads();   // publish every wave's copies before reading
#else
        __syncthreads();
        // ---- fallback staging: global -> VGPR -> LDS ----
        const float* __restrict__ src = xb + (size_t)ch * CHUNK * DDIM;
        #pragma unroll
        for (int j = 0; j < CP_PER_THREAD; ++j) {
            const int idx = j * 256 + tid;
            const int r   = idx >> 5;
            const int c4  = idx & 31;
            const float4 v = ((const float4*)src)[idx];
            float* dst = &lds[cur][r * LDS_STRIDE + c4 * 4];
            dst[0] = v.x; dst[1] = v.y; dst[2] = v.z; dst[3] = v.w;
        }
        __syncthreads();
#endif

        const float* __restrict__ buf = lds[cur];

        // ---- fold this chunk into the column sums (conflict-free reads) ----
        #pragma unroll 8
        for (int r = 0; r < CHUNK / 2; ++r)
            csum += buf[(sh + r) * LDS_STRIDE + sc];

        // ---- WMMA: G += A(16x4) x B(4x16) over CHUNK/4 K-steps ----
        // A[m][k] = X[k][m] (transposed read), B[k][n] = X[k][n].
        #pragma unroll
        for (int k0 = 0; k0 < CHUNK; k0 += 4) {
            const float* row0 = &buf[(k0 + koff) * LDS_STRIDE];
            const float* row1 = &buf[(k0 + koff + 1) * LDS_STRIDE];
            v2f a;
            a.x = row0[mrow + l15];
            a.y = row1[mrow + l15];
            #pragma unroll
            for (int j = 0; j < 8; ++j) {
                v2f bb;
                bb.x = row0[j * 16 + l15];
                bb.y = row1[j * 16 + l15];
                acc[j] = __builtin_amdgcn_wmma_f32_16x16x4_f32(
                    /*neg_a=*/false, a, /*neg_b=*/false, bb,
                    /*c_mod=*/(short)0, acc[j],
                    /*reuse_a=*/false, /*reuse_b=*/false);
            }
        }

        __syncthreads();   // all waves done with buf before it is re-filled
    }

    // ---- partial column sums: combine the two per-column halves in LDS ----
    float* red = (float*)lds;
    red[tid] = csum;
    __syncthreads();
    if (tid < DDIM)
        sum_ws[((size_t)b * KSLICES + s) * DDIM + tid] = red[tid] + red[tid + DDIM];

    // ---- store partial Gram (each element written exactly once) ----
    // C/D layout: VGPR v, lanes 0-15 -> M = v, lanes 16-31 -> M = v + 8.
    float* __restrict__ g = gram_ws + ((size_t)b * KSLICES + s) * (DDIM * DDIM);
    const int mhalf = (lane >> 4) << 3;   // 0 or 8
    #pragma unroll
    for (int j = 0; j < 8; ++j) {
        const int c = j * 16 + l15;
        #pragma unroll
        for (int v = 0; v < 8; ++v) {
            const int r = mrow + v + mhalf;
            g[r * DDIM + c] = acc[j][v];
        }
    }
}

// ---------------------------------------------------------------------------
// Stage 2: fixed-order slice reduction, mean correction, +lam*I, lex row sort.
// One block per batch, 256 threads. Dynamic LDS: cov[128*129] + colsum + pos.
// ---------------------------------------------------------------------------
__global__ __launch_bounds__(256) void finalize_sort(
    const float* __restrict__ gram_ws,
    const float* __restrict__ sum_ws,
    float* __restrict__ out)
{
    extern __shared__ float smem[];
    float* cov    = smem;                       // 128 * 129 (row pad -> no bank conflicts)
    float* colsum = smem + DDIM * (DDIM + 1);   // 128
    int*   pos    = (int*)(colsum + DDIM);      // 128 (rank -> row index)

    const int b   = blockIdx.x;
    const int tid = threadIdx.x;

    if (tid < DDIM) {
        float t = 0.f;
        for (int s = 0; s < KSLICES; ++s)
            t += sum_ws[((size_t)b * KSLICES + s) * DDIM + tid];
        colsum[tid] = t;
    }
    __syncthreads();

    const float invN   = 1.f / (float)NPTS;
    const float invNm1 = 1.f / (float)(NPTS - 1);
    for (int idx = tid; idx < DDIM * DDIM; idx += 256) {
        const int r = idx >> 7, c = idx & (DDIM - 1);
        float gsum = 0.f;
        for (int s = 0; s < KSLICES; ++s)     // fixed order -> deterministic
            gsum += gram_ws[((size_t)b * KSLICES + s) * (DDIM * DDIM) + idx];
        float cv = (gsum - colsum[r] * colsum[c] * invN) * invNm1;
        if (r == c) cv += LAM;
        cov[r * (DDIM + 1) + c] = cv;
    }
    __syncthreads();

    // Lexicographic rank of each row (primary key = column 0), stable by index.
    if (tid < DDIM) {
        const float* myrow = &cov[tid * (DDIM + 1)];
        int rank = 0;
        for (int s = 0; s < DDIM; ++s) {
            const float* srow = &cov[s * (DDIM + 1)];
            bool less = false, decided = false;
            for (int e = 0; e < DDIM; ++e) {
                const float a = srow[e], m = myrow[e];
                if (a != m) { less = (a < m); decided = true; break; }
            }
            if (!decided) less = (s < tid);    // tie-break: original index
            rank += less ? 1 : 0;
        }
        pos[rank] = tid;                       // strict total order -> permutation
    }
    __syncthreads();

    // Coalesced write of rows in sorted order.
    float* __restrict__ ob = out + (size_t)b * (DDIM * DDIM);
    for (int idx = tid; idx < DDIM * DDIM; idx += 256) {
        const int o = idx >> 7, c = idx & (DDIM - 1);
        ob[idx] = cov[pos[o] * (DDIM + 1) + c];
    }
}

// ---------------------------------------------------------------------------
extern "C" void kernel_launch(void* const* d_in, const int* in_sizes, int n_in,
                              void* d_out, int out_size, void* d_ws, size_t ws_size,
                              hipStream_t stream)
{
    (void)in_sizes; (void)n_in; (void)out_size; (void)ws_size;

    const float* x = (const float*)d_in[0];
    float* out = (float*)d_out;

    // Workspace layout (~16.9 MB):
    //   gram_ws: B * KSLICES * 128 * 128 floats = 16 MiB
    //   sum_ws : B * KSLICES * 128 floats       = 128 KiB
    float* gram_ws = (float*)d_ws;
    float* sum_ws  = gram_ws + (size_t)BATCH * KSLICES * DDIM * DDIM;

    gram_partial<<<dim3(BATCH * KSLICES), dim3(256), 0, stream>>>(x, gram_ws, sum_ws);

    const size_t shmem = ((size_t)DDIM * (DDIM + 1) + DDIM + DDIM) * sizeof(float);
    finalize_sort<<<dim3(BATCH), dim3(256), shmem, stream>>>(gram_ws, sum_ws, out);
}